// GCN_17231408791577
// MI455X (gfx1250) — compile-verified
//
#include <hip/hip_runtime.h>

#define N_NODES 100000
#define N_EDGES 1600000
#define C_HID   128
#define C_OUT   47
#define C_OUT_PAD 48

typedef __attribute__((ext_vector_type(2))) float v2f;
typedef __attribute__((ext_vector_type(8))) float v8f;

// ---------------------------------------------------------------- utilities
__global__ void zero_f4(float4* __restrict__ p, int n4) {
    int i = blockIdx.x * blockDim.x + threadIdx.x;
    if (i < n4) p[i] = make_float4(0.f, 0.f, 0.f, 0.f);
}

__global__ void deg_kernel(const long long* __restrict__ rows,
                           float* __restrict__ deg, int ne) {
    int i = blockIdx.x * blockDim.x + threadIdx.x;
    if (i < ne) atomicAdd(&deg[(int)rows[i]], 1.0f);
}

__global__ void rsqrt_kernel(float* __restrict__ d, int n) {
    int i = blockIdx.x * blockDim.x + threadIdx.x;
    if (i < n) d[i] = rsqrtf(d[i] + 1.0f);   // deg >= 0, +1 self-loop
}

// ---------------------------------------------------------------- WMMA GEMM
// H[nrows x ncols] = X[nrows x 128] @ W[128 x ncols], fp32 via v_wmma_f32_16x16x4_f32.
// Block: 256 threads (8 waves). Per block: one 16-row M tile of X in LDS (8 KB)
// plus the ENTIRE W, zero-padded to 128 cols, in LDS (64 KB). Each wave owns one
// 16-col N tile; K loop = 128 in steps of 4 -> 32 WMMA, branch-free inner loop.
__global__ __launch_bounds__(256) void gemm_wmma_f32(
    const float* __restrict__ X, const float* __restrict__ W,
    float* __restrict__ H, int nrows, int ncols, int ldh)
{
    __shared__ float xs[16 * 128];     // X tile
    __shared__ float ws[128 * 128];    // W, columns >= ncols zero-padded
    const int r0  = blockIdx.x * 16;
    const int tid = threadIdx.x;

    // Stage W (coalesced: consecutive threads -> consecutive columns of a row).
    for (int i = tid; i < 128 * 128; i += 256) {
        int kk = i >> 7;          // 0..127 (K row)
        int nn = i & 127;         // 0..127 (N col)
        ws[i] = (nn < ncols) ? W[(size_t)kk * ncols + nn] : 0.f;
    }
    // Stage the 16x128 X tile (zero-pad rows past nrows).
    for (int i = tid; i < 16 * 32; i += 256) {
        int rr = i >> 5;
        int c4 = i & 31;
        float4 v = make_float4(0.f, 0.f, 0.f, 0.f);
        int grow = r0 + rr;
        if (grow < nrows) v = *(const float4*)(X + (size_t)grow * 128 + c4 * 4);
        *(float4*)(xs + rr * 128 + c4 * 4) = v;
    }
    __syncthreads();

    const int wave = tid >> 5;        // 0..7
    const int lane = tid & 31;
    const int half = lane >> 4;       // lane-half selects K sub-pair
    const int l16  = lane & 15;
    const int n    = wave * 16 + l16; // 0..127, always in-bounds of padded ws

    v8f acc = {};
    #pragma unroll
    for (int k = 0; k < 128; k += 4) {
        // A 16x4 fp32: lanes 0-15 -> K=k..k+1, lanes 16-31 -> K=k+2..k+3
        v2f a, b;
        a.x = xs[l16 * 128 + k + 2 * half];
        a.y = xs[l16 * 128 + k + 2 * half + 1];
        // B 4x16 fp32: N striped across lanes, lane-half selects K pair
        b.x = ws[(k + 2 * half)     * 128 + n];
        b.y = ws[(k + 2 * half + 1) * 128 + n];
        acc = __builtin_amdgcn_wmma_f32_16x16x4_f32(
                  false, a, false, b, (short)0, acc, false, false);
    }

    // C/D 16x16 fp32: VGPR r -> M = r + 8*lanehalf, N = lane&15
    if (n < ncols) {
        const int mOff = half * 8;
        #pragma unroll
        for (int r = 0; r < 8; ++r) {
            int row = r0 + r + mOff;
            if (row < nrows) H[(size_t)row * ldh + n] = acc[r];
        }
    }
}

// ---------------------------------------------------------------- scatter (message passing)
// agg[row] += dis[row]*dis[col] * h[col], 128 channels: one wave per edge, float4 per lane.
__global__ void scatter128(const long long* __restrict__ rows,
                           const long long* __restrict__ cols,
                           const float* __restrict__ dis,
                           const float* __restrict__ h,
                           float* __restrict__ agg, int ne)
{
    int t = blockIdx.x * blockDim.x + threadIdx.x;
    int e = t >> 5;
    if (e >= ne) return;
    int g = t & 31;
    int r = (int)rows[e], c = (int)cols[e];
    float nrm = dis[r] * dis[c];
    float4 v = *(const float4*)(h + (size_t)c * 128 + g * 4);
    float* dst = agg + (size_t)r * 128 + g * 4;
    atomicAdd(dst + 0, v.x * nrm);
    atomicAdd(dst + 1, v.y * nrm);
    atomicAdd(dst + 2, v.z * nrm);
    atomicAdd(dst + 3, v.w * nrm);
}

// 47-channel scatter (stride 48). 64 threads per edge, lanes >= 47 idle.
__global__ void scatter_out(const long long* __restrict__ rows,
                            const long long* __restrict__ cols,
                            const float* __restrict__ dis,
                            const float* __restrict__ h,
                            float* __restrict__ agg, int ne)
{
    int t = blockIdx.x * blockDim.x + threadIdx.x;
    int e = t >> 6;
    if (e >= ne) return;
    int c = t & 63;
    if (c >= C_OUT) return;
    int r = (int)rows[e], cc = (int)cols[e];
    float nrm = dis[r] * dis[cc];
    atomicAdd(agg + (size_t)r * C_OUT_PAD + c, nrm * h[(size_t)cc * C_OUT_PAD + c]);
}

// ---------------------------------------------------------------- combine: agg + self_norm*h + b (+relu)
__global__ void combine128(const float* __restrict__ agg, const float* __restrict__ h,
                           const float* __restrict__ dis, const float* __restrict__ b,
                           float* __restrict__ out, int nn, int relu)
{
    int t = blockIdx.x * blockDim.x + threadIdx.x;
    int i = t >> 7;
    if (i >= nn) return;
    int c = t & 127;
    float sn = dis[i] * dis[i];
    size_t idx = (size_t)i * 128 + c;
    float v = agg[idx] + sn * h[idx] + b[c];
    if (relu) v = fmaxf(v, 0.f);
    out[idx] = v;
}

__global__ void combine_out(const float* __restrict__ agg, const float* __restrict__ h,
                            const float* __restrict__ dis, const float* __restrict__ b,
                            float* __restrict__ out, int nn)
{
    int t = blockIdx.x * blockDim.x + threadIdx.x;
    int i = t >> 6;
    if (i >= nn) return;
    int c = t & 63;
    if (c >= C_OUT) return;
    float sn = dis[i] * dis[i];
    size_t pidx = (size_t)i * C_OUT_PAD + c;
    out[(size_t)i * C_OUT + c] = agg[pidx] + sn * h[pidx] + b[c];
}

// ---------------------------------------------------------------- launch
extern "C" void kernel_launch(void* const* d_in, const int* in_sizes, int n_in,
                              void* d_out, int out_size, void* d_ws, size_t ws_size,
                              hipStream_t stream) {
    const float*     x    = (const float*)d_in[0];
    const long long* ei   = (const long long*)d_in[1];   // int64, shape (2, E)
    // d_in[2] = edge_attr (unused by reference)
    const float*     W1   = (const float*)d_in[3];
    const float*     b1   = (const float*)d_in[4];
    const float*     W2   = (const float*)d_in[5];
    const float*     b2   = (const float*)d_in[6];
    const float*     W3   = (const float*)d_in[7];
    const float*     b3   = (const float*)d_in[8];
    float*           out  = (float*)d_out;

    const long long* rows = ei;
    const long long* cols = ei + N_EDGES;

    // Workspace carve-up (floats): dis | hA | hB | agg   (~154 MB total)
    float* dis = (float*)d_ws;
    float* hA  = dis + ((N_NODES + 255) & ~255);
    float* hB  = hA + (size_t)N_NODES * 128;
    float* agg = hB + (size_t)N_NODES * 128;

    const int B = 256;
    auto blocks = [](long long n, int b) { return (unsigned)((n + b - 1) / b); };

    // --- degree / normalization ---
    zero_f4<<<blocks(N_NODES / 4, B), B, 0, stream>>>((float4*)dis, N_NODES / 4);
    deg_kernel<<<blocks(N_EDGES, B), B, 0, stream>>>(rows, dis, N_EDGES);
    rsqrt_kernel<<<blocks(N_NODES, B), B, 0, stream>>>(dis, N_NODES);

    const unsigned gemm_grid = (N_NODES + 15) / 16;        // 6250 blocks

    // --- layer 1: h = relu(GCN(x, W1, b1)) ---
    gemm_wmma_f32<<<gemm_grid, B, 0, stream>>>(x, W1, hA, N_NODES, C_HID, C_HID);
    zero_f4<<<blocks((long long)N_NODES * 32, B), B, 0, stream>>>((float4*)agg, N_NODES * 32);
    scatter128<<<blocks((long long)N_EDGES * 32, B), B, 0, stream>>>(rows, cols, dis, hA, agg, N_EDGES);
    combine128<<<blocks((long long)N_NODES * 128, B), B, 0, stream>>>(agg, hA, dis, b1, hB, N_NODES, 1);

    // --- layer 2 ---
    gemm_wmma_f32<<<gemm_grid, B, 0, stream>>>(hB, W2, hA, N_NODES, C_HID, C_HID);
    zero_f4<<<blocks((long long)N_NODES * 32, B), B, 0, stream>>>((float4*)agg, N_NODES * 32);
    scatter128<<<blocks((long long)N_EDGES * 32, B), B, 0, stream>>>(rows, cols, dis, hA, agg, N_EDGES);
    combine128<<<blocks((long long)N_NODES * 128, B), B, 0, stream>>>(agg, hA, dis, b2, hB, N_NODES, 1);

    // --- layer 3 (47 outputs, stride-48 padded scratch) ---
    gemm_wmma_f32<<<gemm_grid, B, 0, stream>>>(hB, W3, hA, N_NODES, C_OUT, C_OUT_PAD);
    zero_f4<<<blocks((long long)N_NODES * 12, B), B, 0, stream>>>((float4*)agg, N_NODES * 12);
    scatter_out<<<blocks((long long)N_EDGES * 64, B), B, 0, stream>>>(rows, cols, dis, hA, agg, N_EDGES);
    combine_out<<<blocks((long long)N_NODES * 64, B), B, 0, stream>>>(agg, hA, dis, b3, out, N_NODES);
}